// net_54228257079474
// MI455X (gfx1250) — compile-verified
//
#include <hip/hip_runtime.h>
#include <math.h>

typedef __attribute__((ext_vector_type(2))) float v2f;
typedef __attribute__((ext_vector_type(8))) float v8f;

__device__ __forceinline__ v8f wmma4f32(v2f a, v2f b, v8f c) {
  // V_WMMA_F32_16X16X4_F32:  D = A(16x4) x B(4x16) + C(16x16), fp32
  return __builtin_amdgcn_wmma_f32_16x16x4_f32(
      /*neg_a=*/false, a, /*neg_b=*/false, b,
      /*c_mod=*/(short)0, c, /*reuse_a=*/false, /*reuse_b=*/false);
}

// sum across the 16 lanes of each half-wave (rows live per-half on CDNA5 WMMA C layout)
__device__ __forceinline__ float half16_sum(float s) {
  s += __shfl_xor(s, 1, 32);
  s += __shfl_xor(s, 2, 32);
  s += __shfl_xor(s, 4, 32);
  s += __shfl_xor(s, 8, 32);
  return s;
}

// ---------------- Layer 0 edge scatter: acc0[dst, j*8+f] += Kw[j,e] * x[src, f]
__global__ void edge0_kernel(const float* __restrict__ x,
                             const int* __restrict__ src,
                             const int* __restrict__ dst,
                             const float* __restrict__ kw0,
                             const float* __restrict__ kw1,
                             float* __restrict__ acc0, int E) {
  unsigned e = blockIdx.x * blockDim.x + threadIdx.x;
  if (e >= (unsigned)E) return;
  const int s = src[e], t = dst[e];
  const float w0 = kw0[e], w1 = kw1[e];
  const float4* xp = (const float4*)(x + (size_t)s * 8);
  const float4 xa = xp[0], xb = xp[1];
  const float xv[8] = {xa.x, xa.y, xa.z, xa.w, xb.x, xb.y, xb.z, xb.w};
  float* a = acc0 + (size_t)t * 16;
#pragma unroll
  for (int k = 0; k < 8; ++k) atomicAdd(a + k, w0 * xv[k]);
#pragma unroll
  for (int k = 0; k < 8; ++k) atomicAdd(a + 8 + k, w1 * xv[k]);
}

// ---------------- MLP0: h1 = l2norm(acc0[N,16] @ W0[16,32] + b0), 16 nodes / wave
__global__ __launch_bounds__(32) void mlp0_kernel(const float* __restrict__ acc0,
                                                  const float* __restrict__ W0,
                                                  const float* __restrict__ b0,
                                                  float* __restrict__ h1) {
  const int m0 = blockIdx.x * 16;
  const int lane = threadIdx.x;
  const int n = lane & 15;
  const int kb = lane >> 4;
  v8f c0 = {}, c1 = {};
#pragma unroll
  for (int kt = 0; kt < 4; ++kt) {
    const int kA = kt * 4 + 2 * kb;
    v2f a, bb0, bb1;
    const float* ap = acc0 + (size_t)(m0 + n) * 16 + kA;
    a.x = ap[0];
    a.y = ap[1];
    bb0.x = W0[(kA + 0) * 32 + n];
    bb0.y = W0[(kA + 1) * 32 + n];
    bb1.x = W0[(kA + 0) * 32 + 16 + n];
    bb1.y = W0[(kA + 1) * 32 + 16 + n];
    c0 = wmma4f32(a, bb0, c0);
    c1 = wmma4f32(a, bb1, c1);
  }
  const float bias0 = b0[n], bias1 = b0[16 + n];
#pragma unroll
  for (int v = 0; v < 8; ++v) {
    float e0 = c0[v] + bias0;
    float e1 = c1[v] + bias1;
    float s = half16_sum(e0 * e0 + e1 * e1);
    const float invn = 1.0f / fmaxf(sqrtf(s), 1e-12f);
    const int row = m0 + v + 8 * kb;
    h1[(size_t)row * 32 + n] = e0 * invn;
    h1[(size_t)row * 32 + 16 + n] = e1 * invn;
  }
}

// ---------------- Layer 1 edge scatter: 4 threads per edge: (j, half) slices of 16 feats
__global__ void edge1_kernel(const float* __restrict__ h1,
                             const int* __restrict__ src,
                             const int* __restrict__ dst,
                             const float* __restrict__ Kw,
                             float* __restrict__ acc1, int E) {
  unsigned tid = blockIdx.x * blockDim.x + threadIdx.x;
  if (tid >= 4u * (unsigned)E) return;
  const int e = tid >> 2;
  const int j = (tid >> 1) & 1;
  const int hh = tid & 1;
  const int s = src[e], t = dst[e];
  const float w = Kw[(size_t)j * E + e];
  const float4* g = (const float4*)(h1 + (size_t)s * 32 + hh * 16);
  const float4 g0 = g[0], g1 = g[1], g2 = g[2], g3 = g[3];
  const float vv[16] = {g0.x, g0.y, g0.z, g0.w, g1.x, g1.y, g1.z, g1.w,
                        g2.x, g2.y, g2.z, g2.w, g3.x, g3.y, g3.z, g3.w};
  float* a = acc1 + (size_t)t * 64 + j * 32 + hh * 16;
#pragma unroll
  for (int k = 0; k < 16; ++k) atomicAdd(a + k, w * vv[k]);
}

// ---------------- MLP1 fused: out = l2norm(relu(acc1@W1+b1)@W2+b2); 16 nodes / 4-wave block
#define TS 132  // LDS row stride (pad 128 -> 132 dwords: conflict-free column reads)
__global__ __launch_bounds__(128) void mlp1_kernel(const float* __restrict__ acc1,
                                                   const float* __restrict__ W1,
                                                   const float* __restrict__ b1,
                                                   const float* __restrict__ W2,
                                                   const float* __restrict__ b2,
                                                   float* __restrict__ out) {
  __shared__ float tsh[16 * TS];
  __shared__ float rowsq[16];
  const int m0 = blockIdx.x * 16;
  const int wave = threadIdx.x >> 5;
  const int lane = threadIdx.x & 31;
  const int n = lane & 15;
  const int kb = lane >> 4;

  // GEMM1: [16x64] @ [64x128]; wave handles two 16-wide N tiles
#pragma unroll
  for (int tt = 0; tt < 2; ++tt) {
    const int n0 = (wave * 2 + tt) * 16;
    v8f c = {};
    for (int kt = 0; kt < 16; ++kt) {
      const int kA = kt * 4 + 2 * kb;
      v2f a, b;
      const float* ap = acc1 + (size_t)(m0 + n) * 64 + kA;
      a.x = ap[0];
      a.y = ap[1];
      b.x = W1[(kA + 0) * 128 + n0 + n];
      b.y = W1[(kA + 1) * 128 + n0 + n];
      c = wmma4f32(a, b, c);
    }
    const float bias = b1[n0 + n];
#pragma unroll
    for (int v = 0; v < 8; ++v) {
      tsh[(v + 8 * kb) * TS + n0 + n] = fmaxf(c[v] + bias, 0.0f);
    }
  }
  if (threadIdx.x < 16) rowsq[threadIdx.x] = 0.0f;
  __syncthreads();

  // GEMM2: [16x128] @ [128x64]; wave handles one 16-wide N tile
  const int n0 = wave * 16;
  v8f c = {};
  for (int kt = 0; kt < 32; ++kt) {
    const int kA = kt * 4 + 2 * kb;
    v2f a, b;
    a.x = tsh[n * TS + kA];
    a.y = tsh[n * TS + kA + 1];
    b.x = W2[(kA + 0) * 64 + n0 + n];
    b.y = W2[(kA + 1) * 64 + n0 + n];
    c = wmma4f32(a, b, c);
  }
  const float bias = b2[n0 + n];
  float cv[8];
#pragma unroll
  for (int v = 0; v < 8; ++v) {
    cv[v] = c[v] + bias;
    float s = half16_sum(cv[v] * cv[v]);
    if (n == 0) atomicAdd(&rowsq[v + 8 * kb], s);  // ds_add_f32, cross-wave combine
  }
  __syncthreads();
#pragma unroll
  for (int v = 0; v < 8; ++v) {
    const int row = v + 8 * kb;
    const float invn = 1.0f / fmaxf(sqrtf(rowsq[row]), 1e-12f);
    out[(size_t)(m0 + row) * 64 + n0 + n] = cv[v] * invn;
  }
}

extern "C" void kernel_launch(void* const* d_in, const int* in_sizes, int n_in,
                              void* d_out, int out_size, void* d_ws, size_t ws_size,
                              hipStream_t stream) {
  const float* x  = (const float*)d_in[0];
  const int*   ei = (const int*)d_in[1];
  const float* Kw = (const float*)d_in[2];
  const float* W0 = (const float*)d_in[3];
  const float* b0 = (const float*)d_in[4];
  const float* W1 = (const float*)d_in[5];
  const float* b1 = (const float*)d_in[6];
  const float* W2 = (const float*)d_in[7];
  const float* b2 = (const float*)d_in[8];

  const int N = in_sizes[0] / 8;   // 200000 (divisible by 16)
  const int E = in_sizes[1] / 2;   // 3200000
  const int* src = ei;
  const int* dst = ei + E;

  // workspace layout: [acc0 N*16][acc1 N*64][h1 N*32]  (~90 MB)
  float* acc0 = (float*)d_ws;
  float* acc1 = acc0 + (size_t)N * 16;
  float* h1   = acc1 + (size_t)N * 64;

  // zero both accumulators in one contiguous async memset (graph-capture safe)
  hipMemsetAsync(d_ws, 0, (size_t)N * 80 * sizeof(float), stream);

  edge0_kernel<<<(E + 255) / 256, 256, 0, stream>>>(x, src, dst, Kw, Kw + E, acc0, E);
  mlp0_kernel<<<N / 16, 32, 0, stream>>>(acc0, W0, b0, h1);
  edge1_kernel<<<(4 * E + 255) / 256, 256, 0, stream>>>(h1, src, dst, Kw, acc1, E);
  mlp1_kernel<<<N / 16, 128, 0, stream>>>(acc1, W1, b1, W2, b2, (float*)d_out);
}